// pose_estimate_loss_batch_71786083386116
// MI455X (gfx1250) — compile-verified
//
#include <hip/hip_runtime.h>

// ---------------------------------------------------------------------------
// pose_estimate_loss: trilinear SDF sample + Huber mean
//   voxels      [B=128, GL=48, GW=20, GH=18] f32   (8.85 MB, L2-resident)
//   pts_centroid[B, N=16384, 3]              f32   (25.2 MB, streamed)
//   height_gt   [B, N]                       f32   ( 8.4 MB, streamed)
//   out         scalar f32
//
// Memory-bound: ~42 MB @ 23.3 TB/s ≈ 2 us floor. Data path:
//  - async-DMA the 69 KB per-batch voxel slab into WGP LDS
//    (GLOBAL_LOAD_ASYNC_TO_LDS_B128, ASYNCcnt), gathers served by LDS
//  - stream points with b96 loads, 512 blocks x 8 wave32 (LDS caps 4 blk/WGP)
//  - deterministic 2-phase reduction; final 512->1 through the matrix pipe
//    with V_WMMA_F32_16X16X4_F32 (exact f32 row-sums)
// ---------------------------------------------------------------------------

typedef float v2f __attribute__((ext_vector_type(2)));
typedef float v4f __attribute__((ext_vector_type(4)));
typedef float v8f __attribute__((ext_vector_type(8)));
typedef int   v4i __attribute__((ext_vector_type(4)));

#define GLd 48
#define GWd 20
#define GHd 18
#define VOX_PER_B (GLd * GWd * GHd)            // 17280 floats = 69120 B
#define BATCH 128
#define NPTS 16384
#define BLOCKS_PER_B 4
#define PTS_PER_BLOCK (NPTS / BLOCKS_PER_B)    // 4096
#define NBLOCKS (BATCH * BLOCKS_PER_B)         // 512 partials
#define THREADS 256
#define TOTAL_PTS ((float)BATCH * (float)NPTS) // 2097152

struct F3 { float x, y, z; };                  // 12-byte point -> b96 load

__device__ __forceinline__ float readlane_f(float v, int lane) {
    return __builtin_bit_cast(float,
        __builtin_amdgcn_readlane(__builtin_bit_cast(int, v), lane));
}

extern "C" __global__ void __launch_bounds__(THREADS)
sdf_huber_partial_kernel(const float* __restrict__ voxels,
                         const float* __restrict__ pts,
                         const float* __restrict__ hgt,
                         float* __restrict__ partials) {
    extern __shared__ float sv[];              // 17280 floats (dynamic LDS)
    __shared__ float red[THREADS];

    const int tid  = threadIdx.x;
    const int b    = blockIdx.x / BLOCKS_PER_B;
    const int part = blockIdx.x % BLOCKS_PER_B;

    const float* __restrict__ vb = voxels + (size_t)b * VOX_PER_B;

    // ---- Stage this batch's voxel slab into LDS -------------------------
#if __has_builtin(__builtin_amdgcn_global_load_async_to_lds_b128)
    {
        // Fire-and-forget 16B DMA transfers: 4320 x b128, no VGPR round-trip.
        __attribute__((address_space(1))) v4i* gsrc =
            (__attribute__((address_space(1))) v4i*)vb;
        __attribute__((address_space(3))) v4i* ldst =
            (__attribute__((address_space(3))) v4i*)sv;
        for (int i = tid; i < VOX_PER_B / 4; i += THREADS) {
            __builtin_amdgcn_global_load_async_to_lds_b128(
                gsrc + i, ldst + i, 0, 0);
        }
#if __has_builtin(__builtin_amdgcn_s_wait_asynccnt)
        __builtin_amdgcn_s_wait_asynccnt(0);
#else
        asm volatile("s_wait_asynccnt 0x0" ::: "memory");
#endif
    }
#else
    {
        // Fallback: vectorized b128 copy through VGPRs.
        const v4f* __restrict__ v4 = (const v4f*)vb;
        v4f* s4 = (v4f*)sv;
        for (int i = tid; i < VOX_PER_B / 4; i += THREADS) s4[i] = v4[i];
    }
#endif
    __syncthreads();

    // ---- Stream points, gather 8 corners from LDS -----------------------
    const int n0 = part * PTS_PER_BLOCK;
    const F3* __restrict__ p3 = (const F3*)pts;
    float acc = 0.0f;

    for (int n = n0 + tid; n < n0 + PTS_PER_BLOCK; n += THREADS) {
        const size_t pi = (size_t)b * NPTS + (size_t)n;
        const F3    p  = p3[pi];
        const float hg = hgt[pi];

        // shift into grid frame
        const float x = p.x + 2.4f;            // LENGTH/2
        const float y = p.y + 1.0f;            // WIDTH/2
        const float z = p.z + 0.5f * hg;       // + height_gt/2

        const float xm = floorf(x / 0.1f);
        const float ym = floorf(y / 0.1f);
        const float zm = floorf(z / 0.1f);

        // fractional position inside cell: tx = (lx+1)/2
        const float tx = (x - xm * 0.1f) * 10.0f;
        const float ty = (y - ym * 0.1f) * 10.0f;
        const float tz = (z - zm * 0.1f) * 10.0f;
        const float sx = 1.0f - tx, sy = 1.0f - ty, sz = 1.0f - tz;

        const int xi0 = (int)fminf(fmaxf(xm,        0.0f), (float)(GLd - 1));
        const int yi0 = (int)fminf(fmaxf(ym,        0.0f), (float)(GWd - 1));
        const int zi0 = (int)fminf(fmaxf(zm,        0.0f), (float)(GHd - 1));
        const int xi1 = (int)fminf(fmaxf(xm + 1.0f, 0.0f), (float)(GLd - 1));
        const int yi1 = (int)fminf(fmaxf(ym + 1.0f, 0.0f), (float)(GWd - 1));
        const int zi1 = (int)fminf(fmaxf(zm + 1.0f, 0.0f), (float)(GHd - 1));

        const int b11 = (xi1 * GWd + yi1) * GHd;
        const int b10 = (xi1 * GWd + yi0) * GHd;
        const int b01 = (xi0 * GWd + yi1) * GHd;
        const int b00 = (xi0 * GWd + yi0) * GHd;

        const float v111 = sv[b11 + zi1];
        const float v110 = sv[b11 + zi0];
        const float v101 = sv[b10 + zi1];
        const float v100 = sv[b10 + zi0];
        const float v011 = sv[b01 + zi1];
        const float v010 = sv[b01 + zi0];
        const float v001 = sv[b00 + zi1];
        const float v000 = sv[b00 + zi0];

        const float sdf =
              v111 * (tx * ty * tz) + v110 * (tx * ty * sz)
            + v101 * (tx * sy * tz) + v100 * (tx * sy * sz)
            + v011 * (sx * ty * tz) + v010 * (sx * ty * sz)
            + v001 * (sx * sy * tz) + v000 * (sx * sy * sz);

        // Huber(delta=1), target 0
        const float ax = fabsf(sdf);
        acc += (ax < 1.0f) ? 0.5f * sdf * sdf : (ax - 0.5f);
    }

    // ---- Block tree-reduction -> one partial per block -------------------
    red[tid] = acc;
    __syncthreads();
    #pragma unroll
    for (int s = THREADS / 2; s > 0; s >>= 1) {
        if (tid < s) red[tid] += red[tid + s];
        __syncthreads();
    }
    if (tid == 0) partials[blockIdx.x] = red[0];
}

// Final reduction: single wave32. Fold 512 partials into 64 lane slots
// (exact f32, fixed order), then sum all 64 slots through the matrix pipe:
// D = A(16x4, slots) x B(4x16, ones) + 0 -> per-row sums in D.
extern "C" __global__ void __launch_bounds__(32)
sdf_huber_final_kernel(const float* __restrict__ partials,
                       float* __restrict__ out) {
    const int l = threadIdx.x;

    float s0 = 0.0f, s1 = 0.0f;
    #pragma unroll
    for (int i = l; i < NBLOCKS; i += 64)       s0 += partials[i];
    #pragma unroll
    for (int i = l + 32; i < NBLOCKS; i += 64)  s1 += partials[i];

    v2f a;  a[0] = s0;   a[1] = s1;            // A: 16x4 f32, 64 slots
    v2f bo; bo[0] = 1.0f; bo[1] = 1.0f;        // B: 4x16 all-ones
    v8f c = {};                                // C = 0

    v8f d = __builtin_amdgcn_wmma_f32_16x16x4_f32(
        /*neg_a=*/false, a, /*neg_b=*/false, bo,
        /*c_mod=*/(short)0, c, /*reuse_a=*/false, /*reuse_b=*/false);

    // lane 0 holds row-sums 0..7 (one per VGPR), lane 16 holds 8..15
    float s = d[0] + d[1] + d[2] + d[3] + d[4] + d[5] + d[6] + d[7];
    const float total = readlane_f(s, 0) + readlane_f(s, 16);

    if (l == 0) out[0] = total * (1.0f / TOTAL_PTS);
}

extern "C" void kernel_launch(void* const* d_in, const int* in_sizes, int n_in,
                              void* d_out, int out_size, void* d_ws, size_t ws_size,
                              hipStream_t stream) {
    (void)in_sizes; (void)n_in; (void)out_size; (void)ws_size;

    const float* voxels = (const float*)d_in[0];
    const float* pts    = (const float*)d_in[1];
    const float* hgt    = (const float*)d_in[2];
    float* out          = (float*)d_out;
    float* partials     = (float*)d_ws;        // NBLOCKS floats

    sdf_huber_partial_kernel<<<NBLOCKS, THREADS, VOX_PER_B * sizeof(float), stream>>>(
        voxels, pts, hgt, partials);
    sdf_huber_final_kernel<<<1, 32, 0, stream>>>(partials, out);
}